// _ECELoss_26611617366060
// MI455X (gfx1250) — compile-verified
//
#include <hip/hip_runtime.h>
#include <hip/hip_bf16.h>

// ECE loss, faithful to reference:
//   conf = rowmax(logits), pred = argmax (first index on ties)
//   bin  = clamp(ceil(conf*10)-1, 0, 9), valid = conf > 0
//   ece  = sum_b |avg_conf_b - avg_acc_b| * count_b / N
//
// Pure HBM-streaming problem (512 MB reads, ~22 us floor at 23.3 TB/s).
// gfx1250 path: GLOBAL_LOAD_ASYNC_TO_LDS_B128 double-buffered per wave32,
// s_wait_asynccnt handshaking, ds_add_f32 LDS bin atomics, deterministic
// two-pass block-partial reduction in d_ws (no init required: every block
// overwrites its own slot).

#define ECE_NBINS 10
#define ECE_C     128
#define ECE_BLOCK 256
#define ECE_WPB   (ECE_BLOCK / 32)   // waves per block
#define ECE_NBLK  1024

#if defined(__gfx1250__) && __has_builtin(__builtin_amdgcn_global_load_async_to_lds_b128)
#define ECE_ASYNC 1
#else
#define ECE_ASYNC 0
#endif

template <int NWAIT>
__device__ __forceinline__ void ece_wait_async() {
#if defined(__gfx1250__)
#if __has_builtin(__builtin_amdgcn_s_wait_asynccnt)
  __builtin_amdgcn_s_wait_asynccnt(NWAIT);
#else
  asm volatile("s_wait_asynccnt %0" :: "i"(NWAIT) : "memory");
#endif
  asm volatile("" ::: "memory");   // keep LDS reads below the wait
#endif
}

#if ECE_ASYNC
// Builtin wants (int4* addrspace(1), int4* addrspace(3), imm, imm) — per the
// round-1 diagnostic the pointee is a GCC vector_size(16) int.
typedef int ece_v4i __attribute__((vector_size(16)));
typedef __attribute__((address_space(1))) ece_v4i* ece_gptr;
typedef __attribute__((address_space(3))) ece_v4i* ece_lptr;

// Each lane copies 16B: wave covers one 512B row in a single async op.
__device__ __forceinline__ void ece_async_row16(const float* g, float* l) {
  __builtin_amdgcn_global_load_async_to_lds_b128(
      (ece_gptr)g,            // generic -> AS1 addrspacecast (+const drop)
      (ece_lptr)l,            // generic -> AS3 addrspacecast (flat -> LDS off)
      /*offset=*/0, /*cpol=*/0);
}
#endif

__device__ __forceinline__ void ece_row_update(float4 v, int lane, int row,
                                               const int* __restrict__ labels,
                                               float* bins) {
  // Per-lane max/argmax of 4 elements (first index on ties: strict >).
  float m = v.x; int mi = lane * 4;
  if (v.y > m) { m = v.y; mi = lane * 4 + 1; }
  if (v.z > m) { m = v.z; mi = lane * 4 + 2; }
  if (v.w > m) { m = v.w; mi = lane * 4 + 3; }
  // Wave32 butterfly; tie-break to the smaller index (== jnp.argmax).
#pragma unroll
  for (int off = 16; off > 0; off >>= 1) {
    float om = __shfl_xor(m, off, 32);
    int   oi = __shfl_xor(mi, off, 32);
    if (om > m || (om == m && oi < mi)) { m = om; mi = oi; }
  }
  if (lane == 0 && m > 0.0f) {          // conf == 0 falls in no bin
    int b = (int)ceilf(m * (float)ECE_NBINS) - 1;
    b = b < 0 ? 0 : (b > ECE_NBINS - 1 ? ECE_NBINS - 1 : b);
    float acc = (mi == labels[row]) ? 1.0f : 0.0f;
    atomicAdd(&bins[b], 1.0f);                      // ds_add_f32
    atomicAdd(&bins[ECE_NBINS + b], m);
    atomicAdd(&bins[2 * ECE_NBINS + b], acc);
  }
}

__global__ __launch_bounds__(ECE_BLOCK)
void ece_partial_kernel(const float* __restrict__ logits,
                        const int* __restrict__ labels,
                        float* __restrict__ partial,
                        int n, int nblk) {
  __shared__ float bins[3 * ECE_NBINS];
#if ECE_ASYNC
  __shared__ __align__(16) float rowbuf[ECE_WPB][2][ECE_C];  // 8 KB double buffer
#endif
  const int tid  = threadIdx.x;
  const int lane = tid & 31;
  const int wib  = tid >> 5;

  if (tid < 3 * ECE_NBINS) bins[tid] = 0.0f;
  __syncthreads();

  const int W  = nblk * ECE_WPB;                 // total waves (row stride)
  const int r0 = blockIdx.x * ECE_WPB + wib;     // this wave's first row

#if ECE_ASYNC
  int p = 0;
  if (r0 < n)
    ece_async_row16(logits + (size_t)r0 * ECE_C + lane * 4,
                    &rowbuf[wib][0][lane * 4]);
  for (int r = r0; r < n; r += W) {
    const int nxt = r + W;
    if (nxt < n) {                                // prefetch next row
      ece_async_row16(logits + (size_t)nxt * ECE_C + lane * 4,
                      &rowbuf[wib][p ^ 1][lane * 4]);
      ece_wait_async<1>();                        // current row landed
    } else {
      ece_wait_async<0>();
    }
    float4 v = *(const float4*)&rowbuf[wib][p][lane * 4];
    p ^= 1;
    ece_row_update(v, lane, r, labels, bins);
  }
#else
  for (int r = r0; r < n; r += W) {
    float4 v = *(const float4*)(logits + (size_t)r * ECE_C + lane * 4);
    ece_row_update(v, lane, r, labels, bins);
  }
#endif

  __syncthreads();
  if (tid < 3 * ECE_NBINS)
    partial[(size_t)blockIdx.x * 32 + tid] = bins[tid];   // overwrite own slot
}

__global__ void ece_final_kernel(const float* __restrict__ partial,
                                 float* __restrict__ out,
                                 int nblk, float invn) {
  const int lane = threadIdx.x;                    // single wave32
  float cnt = 0.f, cs = 0.f, as = 0.f;
  if (lane < ECE_NBINS) {
    for (int b = 0; b < nblk; ++b) {
      const float* pp = partial + (size_t)b * 32;
      cnt += pp[lane];
      cs  += pp[ECE_NBINS + lane];
      as  += pp[2 * ECE_NBINS + lane];
    }
  }
  float term = 0.f;
  if (lane < ECE_NBINS && cnt > 0.f)
    term = fabsf(cs / cnt - as / cnt) * (cnt * invn);
#pragma unroll
  for (int off = 16; off > 0; off >>= 1)
    term += __shfl_xor(term, off, 32);
  if (lane == 0) out[0] = term;
}

extern "C" void kernel_launch(void* const* d_in, const int* in_sizes, int n_in,
                              void* d_out, int out_size, void* d_ws, size_t ws_size,
                              hipStream_t stream) {
  const float* logits = (const float*)d_in[0];
  const int*   labels = (const int*)d_in[1];
  const int    n      = in_sizes[1];              // number of rows (== |labels|)

  int nblk = ECE_NBLK;
  const size_t slot = 32 * sizeof(float);
  if (ws_size < (size_t)nblk * slot) nblk = (int)(ws_size / slot);
  if (nblk < 1) nblk = 1;

  float* partial = (float*)d_ws;
  ece_partial_kernel<<<nblk, ECE_BLOCK, 0, stream>>>(logits, labels, partial, n, nblk);
  ece_final_kernel<<<1, 32, 0, stream>>>(partial, (float*)d_out, nblk,
                                         1.0f / (float)n);
}